// HAttentionNetwork_17746804867961
// MI455X (gfx1250) — compile-verified
//
#include <hip/hip_runtime.h>
#include <hip/hip_bf16.h>

typedef __attribute__((ext_vector_type(2))) float v2f;
typedef __attribute__((ext_vector_type(8))) float v8f;

#define HIDDEN 256
#define NCLS   53
#define KDIM   512   // HIER * HIDDEN

// ---------------------------------------------------------------------------
// Kernel 1: per-sentence attention logits. One wave32 per sentence.
// lg[i] = ( <rel_emb0[coarse], x[i]>, <rel_emb1[fine], x[i]> )
// ---------------------------------------------------------------------------
__global__ __launch_bounds__(256) void logits_kernel(
    const float* __restrict__ x, const float* __restrict__ re0,
    const float* __restrict__ re1, const int* __restrict__ rlv,
    const int* __restrict__ lbl, float2* __restrict__ lg, int nsent)
{
    int sent = (int)((blockIdx.x * blockDim.x + threadIdx.x) >> 5);
    int lane = threadIdx.x & 31;
    if (sent >= nsent) return;

    int label = lbl[sent];
    int c0 = rlv[2 * label + 0];
    int c1 = rlv[2 * label + 1];

    const float4* xr = (const float4*)(x  + (size_t)sent * HIDDEN);
    const float4* r0 = (const float4*)(re0 + (size_t)c0  * HIDDEN);
    const float4* r1 = (const float4*)(re1 + (size_t)c1  * HIDDEN);

    // 64 float4 per row; lane j takes j and j+32 (coalesced 512B per load).
    float4 xa = xr[lane], xb = xr[lane + 32];
    float4 a0 = r0[lane], b0 = r0[lane + 32];
    float4 a1 = r1[lane], b1 = r1[lane + 32];

    float p0 = xa.x*a0.x + xa.y*a0.y + xa.z*a0.z + xa.w*a0.w
             + xb.x*b0.x + xb.y*b0.y + xb.z*b0.z + xb.w*b0.w;
    float p1 = xa.x*a1.x + xa.y*a1.y + xa.z*a1.z + xa.w*a1.w
             + xb.x*b1.x + xb.y*b1.y + xb.z*b1.z + xb.w*b1.w;

    #pragma unroll
    for (int off = 16; off > 0; off >>= 1) {
        p0 += __shfl_xor(p0, off, 32);
        p1 += __shfl_xor(p1, off, 32);
    }
    if (lane == 0) lg[sent] = make_float2(p0, p1);
}

// ---------------------------------------------------------------------------
// Kernel 2: fused ragged softmax + attention-weighted reduction per bag.
// One 256-thread block per bag; thread t owns hidden dim t.
// stack[b, h]       = sum_i score0_i * x[i,h]
// stack[b, 256 + h] = sum_i score1_i * x[i,h]
// ---------------------------------------------------------------------------
__global__ __launch_bounds__(256) void bag_kernel(
    const float* __restrict__ x, const float2* __restrict__ lg,
    const int* __restrict__ scope, float* __restrict__ stack)
{
    int b   = blockIdx.x;
    int beg = scope[b], end = scope[b + 1];
    int t   = threadIdx.x;

    __shared__ float red0[256], red1[256];
    __shared__ float w0s[64], w1s[64];
    __shared__ float s_m0, s_m1, s_s0, s_s1;

    // ---- max over bag (per level) ----
    float m0 = -INFINITY, m1 = -INFINITY;
    for (int i = beg + t; i < end; i += 256) {
        float2 v = lg[i];
        m0 = fmaxf(m0, v.x);
        m1 = fmaxf(m1, v.y);
    }
    red0[t] = m0; red1[t] = m1;
    __syncthreads();
    for (int s = 128; s > 0; s >>= 1) {
        if (t < s) {
            red0[t] = fmaxf(red0[t], red0[t + s]);
            red1[t] = fmaxf(red1[t], red1[t + s]);
        }
        __syncthreads();
    }
    if (t == 0) { s_m0 = red0[0]; s_m1 = red1[0]; }
    __syncthreads();
    m0 = s_m0; m1 = s_m1;

    // ---- sum of exp ----
    float e0 = 0.f, e1 = 0.f;
    for (int i = beg + t; i < end; i += 256) {
        float2 v = lg[i];
        e0 += expf(v.x - m0);
        e1 += expf(v.y - m1);
    }
    red0[t] = e0; red1[t] = e1;
    __syncthreads();
    for (int s = 128; s > 0; s >>= 1) {
        if (t < s) {
            red0[t] += red0[t + s];
            red1[t] += red1[t + s];
        }
        __syncthreads();
    }
    if (t == 0) { s_s0 = red0[0]; s_s1 = red1[0]; }
    __syncthreads();
    float inv0 = (s_s0 > 0.f) ? 1.f / s_s0 : 0.f;
    float inv1 = (s_s1 > 0.f) ? 1.f / s_s1 : 0.f;

    // ---- weighted accumulation (coalesced over hidden dim) ----
    float acc0 = 0.f, acc1 = 0.f;
    for (int i0 = beg; i0 < end; i0 += 64) {
        int cnt = min(64, end - i0);
        __syncthreads();
        if (t < cnt) {
            float2 v = lg[i0 + t];
            w0s[t] = expf(v.x - m0) * inv0;
            w1s[t] = expf(v.y - m1) * inv1;
        }
        __syncthreads();
        for (int j = 0; j < cnt; ++j) {
            float xv = x[(size_t)(i0 + j) * HIDDEN + t];
            acc0 = fmaf(w0s[j], xv, acc0);
            acc1 = fmaf(w1s[j], xv, acc1);
        }
    }
    stack[(size_t)b * KDIM + t]          = acc0;
    stack[(size_t)b * KDIM + HIDDEN + t] = acc1;
}

// ---------------------------------------------------------------------------
// Kernel 3: out[4096,53] = stack[4096,512] @ disc.T[512,53] + bias  (f32 WMMA)
// 16-row M tile per block; 4 waves per block each own one 16-col N tile.
// V_WMMA_F32_16X16X4_F32 fragment layout:
//   A (16x4): lane&15 = M; vgpr {0,1} = K {0,1}; lanes 16..31 carry K {2,3}
//   B (4x16): lane&15 = N; vgpr {0,1} = K {0,1}; lanes 16..31 carry K {2,3}
//   C (16x16): vgpr r, lanes 0..15 -> M=r,   N=lane
//              vgpr r, lanes 16..31 -> M=r+8, N=lane-16
// Padding columns (n >= 53) read disc row 0; their garbage stays confined to
// output columns that are never stored, so no B masking is needed -> the
// inner loop is pure load + wmma.
// ---------------------------------------------------------------------------
__global__ __launch_bounds__(128) void gemm_kernel(
    const float* __restrict__ S, const float* __restrict__ disc,
    const float* __restrict__ bias, float* __restrict__ out)
{
    int mbase = blockIdx.x * 16;
    int wave  = threadIdx.x >> 5;     // 0..3 -> N tile
    int lane  = threadIdx.x & 31;
    int half  = lane >> 4;            // 0: K 0,1   1: K 2,3
    int lm    = lane & 15;

    int n      = wave * 16 + lm;
    bool nvld  = (n < NCLS);
    int nc     = nvld ? n : 0;        // clamp pad lanes to a finite row

    const float* arow = S    + (size_t)(mbase + lm) * KDIM + 2 * half;
    const float* brow = disc + (size_t)nc           * KDIM + 2 * half;

    v8f acc = {};
    for (int k = 0; k < KDIM; k += 4) {
        v2f a, bb;
        a.x  = arow[k];
        a.y  = arow[k + 1];
        bb.x = brow[k];
        bb.y = brow[k + 1];
        // 8 args: (neg_a, A, neg_b, B, c_mod, C, reuse_a, reuse_b)
        acc = __builtin_amdgcn_wmma_f32_16x16x4_f32(
                  false, a, false, bb, (short)0, acc, false, false);
    }

    if (nvld) {
        float bi = bias[n];
        #pragma unroll
        for (int r = 0; r < 8; ++r) {
            int m = mbase + r + 8 * half;
            out[(size_t)m * NCLS + n] = acc[r] + bi;
        }
    }
}

// ---------------------------------------------------------------------------
extern "C" void kernel_launch(void* const* d_in, const int* in_sizes, int n_in,
                              void* d_out, int out_size, void* d_ws, size_t ws_size,
                              hipStream_t stream) {
    const float* x    = (const float*)d_in[0];   // [N, 256]
    const float* re0  = (const float*)d_in[1];   // [14, 256]
    const float* re1  = (const float*)d_in[2];   // [53, 256]
    const float* disc = (const float*)d_in[3];   // [53, 512]
    const float* bias = (const float*)d_in[4];   // [53]
    const int*   rlv  = (const int*)d_in[5];     // [53, 2]
    const int*   lbl  = (const int*)d_in[6];     // [N]
    const int*   scope= (const int*)d_in[7];     // [B+1]

    int nsent = in_sizes[0] / HIDDEN;            // 262144
    int nbags = in_sizes[7] - 1;                 // 4096

    float2* lg   = (float2*)d_ws;
    float* stack = (float*)((char*)d_ws + (size_t)nsent * sizeof(float2));

    // 8 sentences (waves) per 256-thread block
    logits_kernel<<<(nsent + 7) / 8, 256, 0, stream>>>(x, re0, re1, rlv, lbl, lg, nsent);
    bag_kernel<<<nbags, 256, 0, stream>>>(x, lg, scope, stack);
    gemm_kernel<<<(nbags + 15) / 16, 128, 0, stream>>>(stack, disc, bias, (float*)d_out);
}